// FilterbankCell_34565896798838
// MI455X (gfx1250) — compile-verified
//
#include <hip/hip_runtime.h>
#include <stdint.h>

typedef float        v2f   __attribute__((ext_vector_type(2)));
typedef float        v8f   __attribute__((ext_vector_type(8)));
typedef unsigned int u32x4 __attribute__((ext_vector_type(4)));
typedef int          i32x4 __attribute__((ext_vector_type(4)));
typedef int          i32x8 __attribute__((ext_vector_type(8)));

#define S_LEN   2048
#define B_LEN   2048
#define G_CNT   2
#define K_ORD   12
#define L_T     16                   // timesteps per s-block (WMMA M dim)
#define N_BLK   (S_LEN / L_T)        // 128 sequential s-blocks
#define B_TILE  16                   // batches per workgroup (WMMA N dim)

#define XROW    36                   // padded floats per batch row in x stage (TDM pad)
#define XBUF_F  (B_TILE * XROW)      // 576 floats per buffer
#define OROW    388                  // padded floats per batch row in out stage
#define OBUF_F  (B_TILE * OROW)      // 6208 floats per buffer

// LDS carve (bytes). Pm (A-power table) aliases obuf: it is dead once the
// per-wave WMMA A-fragments are built, before the first obuf write.
#define LDS_OBUF   0
#define LDS_PM     0
#define LDS_HCO    (2 * OBUF_F * 4)                  // 49664
#define LDS_XBUF   (LDS_HCO + 2 * 16 * K_ORD * 4)    // 51200
#define LDS_SBUF   (LDS_XBUF + 2 * XBUF_F * 4)       // 55808
#define LDS_TOTAL  (LDS_SBUF + 2 * G_CNT * 16 * 16 * 4) // 59904 (< 64KB)

// ---- Tensor Data Mover: 2D tile load, 16 rows (batch) x 32 floats (16t x 2g),
// ---- row stride S*G floats, LDS padding 1 DWORD per 8 DWORDs (row stride 36).
__device__ __forceinline__ void tdm_load_x(uint64_t ga, uint32_t lds_off) {
  u32x4 g0;
  g0.x = 1u;                                    // count=1, user descriptor
  g0.y = lds_off;                               // LDS byte address
  g0.z = (uint32_t)ga;                          // global addr [31:0]
  g0.w = ((uint32_t)(ga >> 32) & 0x01FFFFFFu)   // global addr [56:32]
         | (2u << 30);                          // type = 2 ("image")
  i32x8 g1;
  g1[0] = (int)((2u << 16)                      // data_size = 4 bytes
              | (1u << 20)                      // pad_enable
              | (2u << 22)                      // pad_interval: every 8 DWORDs
              | (0u << 25));                    // pad_amount: 1 DWORD
  g1[1] = 0;                                    // tensor_dim0[15:0] (dim0 = 1<<30)
  g1[2] = (int)(((1u << 30) >> 16) & 0xFFFFu);  // tensor_dim0[31:16]; dim1 lo16 = 0
  g1[3] = (int)((((1u << 20) >> 16) & 0xFFFFu)  // tensor_dim1[31:16] (dim1 = 1<<20)
              | (32u << 16));                   // tile_dim0 = 32 elements
  g1[4] = 16;                                   // tile_dim1 = 16, tile_dim2 = 0
  g1[5] = (int)(S_LEN * G_CNT);                 // tensor_dim0_stride lo32 = 4096
  g1[6] = 0;
  g1[7] = 0;
  i32x4 z4; z4[0] = 0; z4[1] = 0; z4[2] = 0; z4[3] = 0;
  i32x8 z8; z8[0] = 0; z8[1] = 0; z8[2] = 0; z8[3] = 0;
  z8[4] = 0; z8[5] = 0; z8[6] = 0; z8[7] = 0;
  __builtin_amdgcn_tensor_load_to_lds(g0, g1, z4, z4, z8, 0);
}

__global__ __launch_bounds__(256, 1)
void laguerre_wmma_scan(const float* __restrict__ x,
                        const float* __restrict__ relax,
                        float* __restrict__ out)
{
  __shared__ __align__(16) unsigned char smem[LDS_TOTAL];
  float* Pm   = (float*)(smem + LDS_PM);    // [g][16][12][12]  A^(d+1)
  float* hco  = (float*)(smem + LDS_HCO);   // [g][16][12]      h[d] = A^d v
  float* xbuf = (float*)(smem + LDS_XBUF);  // [buf][b*36 + padded(t*2+g)]
  float* sbuf = (float*)(smem + LDS_SBUF);  // [buf][g][16][16] carry state
  float* obuf = (float*)(smem + LDS_OBUF);  // [buf][b*388 + t*24 + g*12 + k]

  const int tid  = threadIdx.x;
  const int wave = tid >> 5;
  const int lane = tid & 31;
  const int half = lane >> 4;
  const int ln16 = lane & 15;
  const int b0   = blockIdx.x * B_TILE;
  const uint32_t lds_base = (uint32_t)(uintptr_t)(void*)smem;

  // ---------------- zero initial carry state (both buffers) ----------------
  for (int i = tid; i < 2 * G_CNT * 16 * 16; i += 256) sbuf[i] = 0.0f;

  // ------------- per-group transition matrix A and input vector v ----------
  for (int gg = 0; gg < G_CNT; ++gg) {
    float rx = relax[gg];
    float sa = sqrtf(rx), sqb = sqrtf(1.0f - rx);
    if (tid < K_ORD * K_ORD) {
      int r = tid / K_ORD, c = tid % K_ORD;
      float a = 0.0f;
      if (c == r)      a = sa;                                       // diag
      else if (c < r)  a = (sa * sa - 1.0f) * powf(sa, (float)(r - c - 1));
      Pm[((gg * 16 + 0) * K_ORD + r) * K_ORD + c] = a;               // A^1
    }
    if (tid < K_ORD)
      hco[(gg * 16 + 0) * K_ORD + tid] = powf(sa, (float)tid) * sqb; // v
  }
  __syncthreads();
  // powers A^(d+1) and impulse response h[d] = A^d v, d = 1..15
  for (int d = 1; d < 16; ++d) {
    for (int gg = 0; gg < G_CNT; ++gg) {
      if (tid < K_ORD * K_ORD) {
        int r = tid / K_ORD, c = tid % K_ORD;
        float s = 0.0f;
        for (int m = 0; m < K_ORD; ++m)
          s += Pm[((gg * 16 + 0) * K_ORD + r) * K_ORD + m] *
               Pm[((gg * 16 + (d - 1)) * K_ORD + m) * K_ORD + c];
        Pm[((gg * 16 + d) * K_ORD + r) * K_ORD + c] = s;
      }
      if (tid < K_ORD) {
        float s = 0.0f;
        for (int m = 0; m < K_ORD; ++m)
          s += Pm[((gg * 16 + 0) * K_ORD + tid) * K_ORD + m] *
               hco[(gg * 16 + (d - 1)) * K_ORD + m];
        hco[(gg * 16 + d) * K_ORD + tid] = s;
      }
    }
    __syncthreads();
  }

  // ------------- per-wave tile assignment + WMMA A-fragments ---------------
  const int g  = wave >> 2;        // 4 waves per group
  const int k0 = (wave & 3) * 3;   // 3 filter orders per wave

  // A-frag (16x4 f32): lane = M (ln16), vgpr q -> K = q + 2*half within slice.
  v2f Wf[3][4], Cf[3][3];
  #pragma unroll
  for (int kk = 0; kk < 3; ++kk) {
    const int k = k0 + kk;
    #pragma unroll
    for (int c = 0; c < 4; ++c)
      #pragma unroll
      for (int q = 0; q < 2; ++q) {
        int i = 4 * c + q + 2 * half;              // time input index
        int d = ln16 - i;                          // Toeplitz: W[j][i] = h[j-i]
        Wf[kk][c][q] = (d >= 0) ? hco[(g * 16 + d) * K_ORD + k] : 0.0f;
      }
    #pragma unroll
    for (int c = 0; c < 3; ++c)
      #pragma unroll
      for (int q = 0; q < 2; ++q) {
        int mm = 4 * c + q + 2 * half;             // carry-state index 0..11
        Cf[kk][c][q] = Pm[((g * 16 + ln16) * K_ORD + k) * K_ORD + mm];
      }
  }

  const float* xg = x   + (uint64_t)b0 * (S_LEN * G_CNT);
  float*     outg = out + (uint64_t)b0 * ((uint64_t)S_LEN * G_CNT * K_ORD);

  // prologue: TDM-prefetch x tile for block 0 into xbuf[0]
  if (wave == 0)
    tdm_load_x((uint64_t)(uintptr_t)xg, lds_base + LDS_XBUF);

  // ------------------------------ scan loop --------------------------------
  for (int blk = 0; blk < N_BLK; ++blk) {
    const int buf = blk & 1;
    const int t0  = blk * L_T;

    // my async flushes from blk-2 done (<=6 outstanding = only blk-1 left)
    asm volatile("s_wait_asynccnt 6" ::: "memory");
    // x tile for this block landed in LDS
    if (wave == 0) __builtin_amdgcn_s_wait_tensorcnt(0);
    __syncthreads();

    // TDM-prefetch next x tile into the other buffer
    if (wave == 0 && blk + 1 < N_BLK)
      tdm_load_x((uint64_t)(uintptr_t)(xg + (size_t)(t0 + L_T) * G_CNT),
                 lds_base + LDS_XBUF + (uint32_t)((buf ^ 1) * XBUF_F * 4));

    // B-frag (4x16): lane = N col (ln16), vgpr q -> K row = q + 2*half.
    const float* xb   = xbuf + buf * XBUF_F;
    const float* sinp = sbuf + ((buf * G_CNT + g) * 16) * 16;
    v2f Xf[4], Sf[3];
    #pragma unroll
    for (int c = 0; c < 4; ++c)
      #pragma unroll
      for (int q = 0; q < 2; ++q) {
        int i = 4 * c + q + 2 * half;
        int e = i * G_CNT + g;
        Xf[c][q] = xb[ln16 * XROW + e + (e >> 3)];   // TDM-padded row
      }
    #pragma unroll
    for (int c = 0; c < 3; ++c)
      #pragma unroll
      for (int q = 0; q < 2; ++q) {
        int s = 4 * c + q + 2 * half;
        Sf[c][q] = sinp[s * 16 + ln16];
      }

    float* ob    = obuf + buf * OBUF_F;
    float* snext = sbuf + (((buf ^ 1) * G_CNT + g) * 16) * 16;
    #pragma unroll
    for (int kk = 0; kk < 3; ++kk) {
      v8f acc = {};
      #pragma unroll
      for (int c = 0; c < 4; ++c)     // Y += W_k * X   (16x16x16 via 4x K=4)
        acc = __builtin_amdgcn_wmma_f32_16x16x4_f32(
                  false, Wf[kk][c], false, Xf[c], (short)0, acc, false, false);
      #pragma unroll
      for (int c = 0; c < 3; ++c)     // Y += C_k * S_in (carry injection)
        acc = __builtin_amdgcn_wmma_f32_16x16x4_f32(
                  false, Cf[kk][c], false, Sf[c], (short)0, acc, false, false);

      const int k = k0 + kk;
      // D layout: vgpr r -> t = r + 8*half, col ln16 = batch. Padded rows
      // (stride 388 floats) make the 16 lanes hit 16 distinct banks.
      #pragma unroll
      for (int r = 0; r < 8; ++r) {
        int t = r + 8 * half;
        ob[ln16 * OROW + t * (G_CNT * K_ORD) + g * K_ORD + k] = acc[r];
      }
      if (half) snext[k * 16 + ln16] = acc[7];       // carry = y[t0+15]
    }

    __syncthreads();   // all D tiles + carries staged

    // async flush: wave w stores batch rows 2w, 2w+1 (384 floats each,
    // contiguous in global: out[b][t0..t0+15][g][k]) via B128 LDS->global.
    {
      const uint32_t obase = lds_base + LDS_OBUF + (uint32_t)(buf * OBUF_F * 4);
      #pragma unroll
      for (int q = 0; q < 6; ++q) {
        int ch = q * 32 + lane;                  // 0..191
        int b  = 2 * wave + (ch >= 96 ? 1 : 0);
        int j  = (ch % 96) * 4;                  // float offset within row
        uint32_t laddr = obase + (uint32_t)(b * OROW + j) * 4u;
        const float* gp = outg + (uint64_t)b * (S_LEN * G_CNT * K_ORD)
                               + (uint64_t)t0 * (G_CNT * K_ORD) + j;
        asm volatile("global_store_async_from_lds_b128 %0, %1, off"
                     :: "v"((uint64_t)(uintptr_t)gp), "v"(laddr)
                     : "memory");
      }
    }
  }

  asm volatile("s_wait_asynccnt 0" ::: "memory");
  if (wave == 0) __builtin_amdgcn_s_wait_tensorcnt(0);
}

extern "C" void kernel_launch(void* const* d_in, const int* in_sizes, int n_in,
                              void* d_out, int out_size, void* d_ws, size_t ws_size,
                              hipStream_t stream) {
  (void)in_sizes; (void)n_in; (void)out_size; (void)d_ws; (void)ws_size;
  const float* x     = (const float*)d_in[0];
  const float* relax = (const float*)d_in[1];
  float*       out   = (float*)d_out;
  laguerre_wmma_scan<<<dim3(B_LEN / B_TILE), dim3(256), 0, stream>>>(x, relax, out);
}